// DenseCapsule_40922448396877
// MI455X (gfx1250) — compile-verified
//
#include <hip/hip_runtime.h>
#include <cstdint>
#include <cstddef>

// ---------------------------------------------------------------------------
// DenseCapsule dynamic routing, fully fused for MI455X (gfx1250, wave32).
//
//   x      : [512, 1152, 8]  f32
//   weight : [10, 1152, 16, 8] f32
//   out    : [512, 10, 16]   f32
//
// Strategy: never materialize x_hat (377 MB). Exploit linearity of routing
// logits: b_i[o,n] = <U_i[o,:], x_hat[b,o,n,:]> with U_i = sum of previous
// outputs. Recompute x_hat tiles each routing pass via V_WMMA_F32_16X16X4_F32
// (W[o,n] 16x8 @ x^T 8x16 -> xhat^T 16 dout x 16 batch), contraction K=8 as
// two K=4 WMMAs. x (19 MB) and W (6 MB) stay L2-resident across the 3 passes.
// ---------------------------------------------------------------------------

namespace {
constexpr int kB    = 512;
constexpr int kNin  = 1152;
constexpr int kDin  = 8;
constexpr int kNout = 10;
constexpr int kDout = 16;

constexpr int kTiles     = kB / 16;          // 32 batch tiles of 16
constexpr int kChunks    = 16;               // n-chunks per tile
constexpr int kNPerChunk = kNin / kChunks;   // 72
constexpr int kWaves     = 8;                // 256 threads = 8 wave32
constexpr int kNPerWave  = kNPerChunk / kWaves; // 9
constexpr int kRec       = kNout * 16 * 16;  // 2560 floats per partial record
constexpr int kUElems    = kB * kNout * kDout; // 81920 floats
}

typedef float v2f __attribute__((ext_vector_type(2)));
typedef float v4f __attribute__((ext_vector_type(4)));
typedef float v8f __attribute__((ext_vector_type(8)));

// ---------------------------------------------------------------------------
// Kernel A: for one (batch-tile, n-chunk) compute the c-weighted partial sum
//   s[o, i, batch] = sum_{n in chunk} c[b,o,n] * x_hat[b,o,n,i]
// where c = softmax_o(<U, x_hat>)  (uniform 0.1 on the first pass, haveU==0).
// ---------------------------------------------------------------------------
__global__ __launch_bounds__(256) void caps_partial(
    const float* __restrict__ x, const float* __restrict__ w,
    const float* __restrict__ U, float* __restrict__ partial, int haveU)
{
  __shared__ __align__(16) float u_lds[16 * kNout * kDout]; // 10 KB
  __shared__ __align__(16) float s_lds[kRec];               // 10 KB

  const int tile  = blockIdx.x;   // 0..31 (batch tile)
  const int chunk = blockIdx.y;   // 0..15 (n chunk)
  const int tid   = threadIdx.x;
  const int lane  = tid & 31;
  const int wave  = tid >> 5;

  // Cooperative load of this tile's U into LDS (zeros on first pass),
  // and zero the cross-wave reduction buffer.
  for (int e = tid; e < 16 * kNout * kDout; e += 256) {
    u_lds[e] = haveU ? U[(size_t)tile * 16 * kNout * kDout + e] : 0.0f;
    s_lds[e] = 0.0f;
  }
  __syncthreads();

  // wave32 WMMA operand addressing (f32 16x16x4):
  //   A 16x4: lanes 0-15 -> M=lane, K={0,1}; lanes 16-31 -> M=lane-16, K={2,3}
  //   B  4x16: lanes 0-15 -> N=lane, K={0,1}; lanes 16-31 -> N=lane-16, K={2,3}
  const int m    = lane & 15;      // A row (dout) / B col (batch-in-tile)
  const int half = lane >> 4;
  const int k0   = half * 2;       // first K index this lane supplies

  const float* __restrict__ xrow =
      x + (size_t)(tile * 16 + m) * (kNin * kDin);  // + n*8 + j

  v8f zero = {};
  v8f s[kNout];
  #pragma unroll
  for (int o = 0; o < kNout; ++o) s[o] = zero;

  const int n0 = chunk * kNPerChunk + wave * kNPerWave;

  for (int nn = 0; nn < kNPerWave; ++nn) {
    const int n = n0 + nn;

    // B operand: x^T [8 din x 16 batch]; lane loads x[b, n, k0..k0+1] etc.
    v2f bx0 = *(const v2f*)(xrow + n * kDin + k0);
    v2f bx1 = *(const v2f*)(xrow + n * kDin + k0 + 4);

    // x_hat^T tiles for all 10 output capsules (needed before softmax)
    v8f xh[kNout];
    #pragma unroll
    for (int o = 0; o < kNout; ++o) {
      const float* wp =
          w + ((size_t)(o * kNin + n) * kDout + m) * kDin + k0;
      v2f a0 = *(const v2f*)(wp);
      v2f a1 = *(const v2f*)(wp + 4);
      v8f acc = zero;
      acc = __builtin_amdgcn_wmma_f32_16x16x4_f32(
          false, a0, false, bx0, (short)0, acc, false, false);
      acc = __builtin_amdgcn_wmma_f32_16x16x4_f32(
          false, a1, false, bx1, (short)0, acc, false, false);
      xh[o] = acc;  // element (i, b): vgpr=i%8, lane=(i/8)*16 + b
    }

    // Routing coefficients for batch row (= lane&15)
    float c[kNout];
    if (haveU) {
      float logit[kNout];
      #pragma unroll
      for (int o = 0; o < kNout; ++o) {
        // This lane holds xh[o] for i = 8*half .. 8*half+7 of batch m.
        const v4f* up =
            (const v4f*)&u_lds[(m * kNout + o) * kDout + 8 * half];
        v4f u0 = up[0], u1 = up[1];
        float p = u0.x * xh[o][0] + u0.y * xh[o][1] +
                  u0.z * xh[o][2] + u0.w * xh[o][3] +
                  u1.x * xh[o][4] + u1.y * xh[o][5] +
                  u1.z * xh[o][6] + u1.w * xh[o][7];
        // SWAPX16 (xor 0x10) combines i=0..7 and i=8..15 halves.
        p += __int_as_float(
            __builtin_amdgcn_ds_swizzle(__float_as_int(p), 0x401F));
        logit[o] = p;
      }
      float mx = logit[0];
      #pragma unroll
      for (int o = 1; o < kNout; ++o) mx = fmaxf(mx, logit[o]);
      float sum = 0.0f;
      #pragma unroll
      for (int o = 0; o < kNout; ++o) {
        c[o] = __expf(logit[o] - mx);
        sum += c[o];
      }
      float inv = __frcp_rn(sum);
      #pragma unroll
      for (int o = 0; o < kNout; ++o) c[o] *= inv;
    } else {
      // softmax of all-zero logits over 10 capsules
      #pragma unroll
      for (int o = 0; o < kNout; ++o) c[o] = 0.1f;
    }

    // s[o] += c[b,o] * xh[o]   (c is uniform across i for this lane's batch)
    #pragma unroll
    for (int o = 0; o < kNout; ++o) {
      #pragma unroll
      for (int v = 0; v < 8; ++v) s[o][v] += c[o] * xh[o][v];
    }
  }

  // Deterministic cross-wave reduction (fixed wave order, no float atomics).
  for (int wv = 0; wv < kWaves; ++wv) {
    if (wave == wv) {
      #pragma unroll
      for (int o = 0; o < kNout; ++o) {
        #pragma unroll
        for (int v = 0; v < 8; ++v) {
          const int i = v + 8 * half;
          s_lds[(o * 16 + i) * 16 + m] += s[o][v];
        }
      }
    }
    __syncthreads();
  }

  // Emit this block's partial record.
  float* __restrict__ rec =
      partial + (size_t)(chunk * kTiles + tile) * kRec;
  for (int e = tid; e < kRec; e += 256) rec[e] = s_lds[e];
}

// ---------------------------------------------------------------------------
// Kernel B: fold the 16 n-chunk partials of one batch tile, apply squash,
// then either accumulate into U (intermediate pass) or write d_out (final).
// ---------------------------------------------------------------------------
__global__ __launch_bounds__(256) void caps_reduce(
    const float* __restrict__ partial, float* __restrict__ U,
    float* __restrict__ out, int first, int isFinal)
{
  __shared__ float s_lds[kRec];
  const int tile = blockIdx.x;
  const int tid  = threadIdx.x;

  for (int e = tid; e < kRec; e += 256) {
    float acc = 0.0f;
    for (int c = 0; c < kChunks; ++c)  // fixed order -> deterministic
      acc += partial[(size_t)(c * kTiles + tile) * kRec + e];
    s_lds[e] = acc;
  }
  __syncthreads();

  if (tid < 16 * kNout) {               // one thread per (o, batch) pair
    const int o = tid >> 4;
    const int b = tid & 15;
    float v[kDout];
    float ss = 0.0f;
    #pragma unroll
    for (int i = 0; i < kDout; ++i) {
      v[i] = s_lds[(o * 16 + i) * 16 + b];
      ss += v[i] * v[i];
    }
    const float norm  = __fsqrt_rn(ss);
    const float scale = ss / ((1.0f + ss) * (norm + 1e-8f));
    const size_t base = ((size_t)(tile * 16 + b) * kNout + o) * kDout;
    if (isFinal) {
      #pragma unroll
      for (int i = 0; i < kDout; ++i) out[base + i] = scale * v[i];
    } else if (first) {
      #pragma unroll
      for (int i = 0; i < kDout; ++i) U[base + i] = scale * v[i];
    } else {
      #pragma unroll
      for (int i = 0; i < kDout; ++i) U[base + i] += scale * v[i];
    }
  }
}

// ---------------------------------------------------------------------------
extern "C" void kernel_launch(void* const* d_in, const int* in_sizes, int n_in,
                              void* d_out, int out_size, void* d_ws,
                              size_t ws_size, hipStream_t stream)
{
  const float* x = (const float*)d_in[0];   // [512,1152,8]
  const float* w = (const float*)d_in[1];   // [10,1152,16,8]
  float* out = (float*)d_out;               // [512,10,16]

  // Workspace: U (accumulated outputs) + chunk partials. ~5.3 MB total.
  float* U       = (float*)d_ws;                 // kUElems floats
  float* partial = U + kUElems;                  // kChunks*kTiles*kRec floats

  const dim3 gridA(kTiles, kChunks);

  // Routing pass 0: c uniform (softmax of zeros), U undefined -> haveU=0.
  caps_partial<<<gridA, 256, 0, stream>>>(x, w, U, partial, 0);
  caps_reduce <<<kTiles, 256, 0, stream>>>(partial, U, out, 1, 0);
  // Routing pass 1: logits = <U, x_hat>.
  caps_partial<<<gridA, 256, 0, stream>>>(x, w, U, partial, 1);
  caps_reduce <<<kTiles, 256, 0, stream>>>(partial, U, out, 0, 0);
  // Routing pass 2 (final): write squashed outputs.
  caps_partial<<<gridA, 256, 0, stream>>>(x, w, U, partial, 1);
  caps_reduce <<<kTiles, 256, 0, stream>>>(partial, U, out, 0, 1);

  (void)in_sizes; (void)n_in; (void)out_size; (void)ws_size;
}